// HCC_71880572666195
// MI455X (gfx1250) — compile-verified
//
#include <hip/hip_runtime.h>
#include <hip/hip_bf16.h>
#include <stdint.h>

// ---------------------------------------------------------------------------
// argmax over 500k x 400 f32 (800 MB stream -> ~34us roofline @ 23.3 TB/s),
// then tiny confusion-matrix / top-k / pair logic. Memory-bound; no MMA
// structure -> CDNA5 path used is the Tensor Data Mover: per-wave
// double-buffered tensor_load_to_lds (12.8 KB 2-D tiles), TENSORcnt sync.
// ---------------------------------------------------------------------------

#define NUM_CLASS      400
#define K_CORR         20
#define ROWS_PER_DMA   8
#define ROW_ELEMS      NUM_CLASS
#define ROW_BYTES      (NUM_CLASS * 4)
#define BUF_BYTES      (ROWS_PER_DMA * ROW_BYTES)      // 12800 B per buffer
#define WAVES_PER_BLOCK 4
#define THREADS_MAIN   (WAVES_PER_BLOCK * 32)          // 128

typedef __attribute__((ext_vector_type(4))) unsigned int u32x4;
typedef __attribute__((ext_vector_type(8))) unsigned int u32x8;

// Issue one TDM load: 2-D tile (tile_dim0 = 400 f32 elems, tile_dim1 = 8 rows)
// from global memory into LDS at byte offset lds_byte_off. rows_in_tensor is
// used as tensor_dim1 so rows past the end of the batch read back as zeros
// (OOB-safe tail). Issued once per wave (TDM ignores EXEC); operands are
// wave-uniform so "s" constraints are legal.
__device__ __forceinline__ void tdm_load_rows(const float* gptr,
                                              unsigned lds_byte_off,
                                              unsigned rows_in_tensor) {
  unsigned long long ga = (unsigned long long)(uintptr_t)gptr;
  u32x4 g0;
  g0[0] = 1u;                                          // count=1 (valid user D#)
  g0[1] = lds_byte_off;                                // lds_addr (dyn-LDS base = 0)
  g0[2] = (unsigned)(ga & 0xFFFFFFFFull);              // global_addr[31:0]
  g0[3] = (unsigned)((ga >> 32) & 0x01FFFFFFull)       // global_addr[56:32]
        | (2u << 30);                                  // type = 2 ("image")
  u32x8 g1;
  g1[0] = (2u << 16);                                  // data_size = 2 -> 4 bytes
  g1[1] = ((unsigned)ROW_ELEMS & 0xFFFFu) << 16;       // tensor_dim0[15:0]
  g1[2] = ((unsigned)ROW_ELEMS >> 16)                  // tensor_dim0[31:16]
        | ((rows_in_tensor & 0xFFFFu) << 16);          // tensor_dim1[15:0]
  g1[3] = (rows_in_tensor >> 16)                       // tensor_dim1[31:16]
        | ((unsigned)ROW_ELEMS << 16);                 // tile_dim0 = 400
  g1[4] = (unsigned)ROWS_PER_DMA;                      // tile_dim1 = 8, tile_dim2 = 0
  g1[5] = (unsigned)ROW_ELEMS;                         // tensor_dim0_stride = 400
  g1[6] = 0u;                                          // stride hi / dim1_stride lo
  g1[7] = 0u;                                          // dim1_stride hi
  // 2-D tensor: VADDR2/VADDR3 omitted (NULL). Portable across both toolchains.
  asm volatile("tensor_load_to_lds %0, %1" :: "s"(g0), "s"(g1) : "memory");
}

__global__ void zero_u32(unsigned* __restrict__ p, int n) {
  int i = blockIdx.x * blockDim.x + threadIdx.x;
  if (i < n) p[i] = 0u;
}

// Streaming kernel: per-wave TDM double-buffer -> LDS argmax -> cm atomicAdd.
__global__ __launch_bounds__(THREADS_MAIN)
void argmax_cm_kernel(const float* __restrict__ scores,
                      const int* __restrict__ y,
                      unsigned* __restrict__ cm,
                      int batch) {
  extern __shared__ float lds[];   // WAVES_PER_BLOCK * 2 * BUF_BYTES bytes
  const int wave = threadIdx.x >> 5;
  const int lane = threadIdx.x & 31;
  const int ngroups = (batch + ROWS_PER_DMA - 1) / ROWS_PER_DMA;
  const int gstride = (int)(gridDim.x * WAVES_PER_BLOCK);
  int g = (int)(blockIdx.x * WAVES_PER_BLOCK + wave);
  const unsigned my_base = (unsigned)wave * 2u * (unsigned)BUF_BYTES;

  // Prologue: issue first tile (clamped so the DMA address is always valid;
  // TDM/SOPP execute regardless of EXEC, so operands must be safe everywhere).
  {
    int gc = (g < ngroups) ? g : 0;
    tdm_load_rows(scores + (size_t)gc * ROW_ELEMS * ROWS_PER_DMA,
                  my_base, (unsigned)(batch - gc * ROWS_PER_DMA));
  }

  int p = 0;
  for (; g < ngroups; g += gstride) {
    // Prefetch next tile into the other buffer (dummy refetch of the current
    // tile if no next work — harmless, keeps TDM issue unconditional).
    int gn = g + gstride;
    int gl = (gn < ngroups) ? gn : g;
    tdm_load_rows(scores + (size_t)gl * ROW_ELEMS * ROWS_PER_DMA,
                  my_base + (unsigned)(p ^ 1) * (unsigned)BUF_BYTES,
                  (unsigned)(batch - gl * ROWS_PER_DMA));
    // TENSORcnt completes in-order per wave: <=1 outstanding => tile g done.
    __builtin_amdgcn_s_wait_tensorcnt(1);

    const float4* buf =
        (const float4*)(lds + (my_base >> 2) + (unsigned)p * (BUF_BYTES / 4));
    for (int r = 0; r < ROWS_PER_DMA; ++r) {
      int row = g * ROWS_PER_DMA + r;
      if (row >= batch) break;
      const float4* rp = buf + r * (ROW_ELEMS / 4);
      float best = -3.4e38f;
      int bi = 0;
      // Ascending-index scan per lane -> strict '>' keeps the first max.
      for (int k = lane; k < ROW_ELEMS / 4; k += 32) {
        float4 v = rp[k];
        int e = 4 * k;
        if (v.x > best) { best = v.x; bi = e;     }
        if (v.y > best) { best = v.y; bi = e + 1; }
        if (v.z > best) { best = v.z; bi = e + 2; }
        if (v.w > best) { best = v.w; bi = e + 3; }
      }
      // Wave32 reduction; ties resolved to lowest index (jnp.argmax semantics).
      for (int off = 16; off > 0; off >>= 1) {
        float ov = __shfl_down(best, off, 32);
        int   oi = __shfl_down(bi,   off, 32);
        if (ov > best || (ov == best && oi < bi)) { best = ov; bi = oi; }
      }
      if (lane == 0) {
        int t = y[row];
        atomicAdd(&cm[(size_t)t * NUM_CLASS + (unsigned)bi], 1u);
      }
    }
    p ^= 1;
  }
  // s_endpgm performs an implicit wait-idle for the trailing dummy DMA.
}

// Tiny epilogue: masked column max, totals/present, top-20, pair emission.
__global__ __launch_bounds__(512)
void finalize_kernel(const unsigned* __restrict__ cm, float* __restrict__ out) {
  __shared__ float entry[NUM_CLASS];
  __shared__ int   eidx[NUM_CLASS];
  __shared__ float totals[NUM_CLASS];
  __shared__ int   present[NUM_CLASS];
  __shared__ float score[NUM_CLASS];
  __shared__ float redv[512];
  __shared__ int   redi[512];
  __shared__ int   ranked[K_CORR];
  __shared__ int   removed[NUM_CLASS];

  const int tid = threadIdx.x;
  for (int c = tid; c < NUM_CLASS; c += 512) {
    totals[c] = 0.0f; present[c] = 0; removed[c] = 0;
  }
  __syncthreads();

  // Per-column max over rows with diagonal masked to 0; tie -> lowest row,
  // all-zero column -> idx 0 (matches masked argmax in the reference).
  if (tid < NUM_CLASS) {
    float best = -1.0f; int bi = 0;
    for (int r = 0; r < NUM_CLASS; ++r) {
      float v = (r == tid) ? 0.0f : (float)cm[(size_t)r * NUM_CLASS + tid];
      if (v > best) { best = v; bi = r; }
    }
    entry[tid] = best; eidx[tid] = bi;
  }
  __syncthreads();

  if (tid < NUM_CLASS) {
    atomicAdd(&totals[eidx[tid]], entry[tid]);   // ds_add_f32
    atomicOr(&present[eidx[tid]], 1);
  }
  __syncthreads();
  if (tid < NUM_CLASS) score[tid] = present[tid] ? totals[tid] : -1.0f;
  __syncthreads();

  // 20x (max, lowest-index-on-tie) extraction == stable argsort(-score)[:20].
  for (int i = 0; i < K_CORR; ++i) {
    redv[tid] = (tid < NUM_CLASS) ? score[tid] : -3.4e38f;
    redi[tid] = tid;
    __syncthreads();
    for (int s = 256; s > 0; s >>= 1) {
      if (tid < s) {
        float ov = redv[tid + s]; int oi = redi[tid + s];
        if (ov > redv[tid] || (ov == redv[tid] && oi < redi[tid])) {
          redv[tid] = ov; redi[tid] = oi;
        }
      }
      __syncthreads();
    }
    if (tid == 0) { ranked[i] = redi[0]; score[redi[0]] = -3.4e38f; }
    __syncthreads();
  }

  // Pair emission: ranked[i] pairs with the (k-i-1) smallest class ids not in
  // ranked[0..i]. 190 pairs, serial on one thread (trivial work).
  if (tid == 0) {
    int pos = 0;
    for (int i = 0; i < K_CORR; ++i) {
      removed[ranked[i]] = 1;
      int n = K_CORR - i - 1, c2 = 0;
      for (int cls = 0; cls < NUM_CLASS && c2 < n; ++cls) {
        if (!removed[cls]) {
          out[pos * 2 + 0] = (float)ranked[i];
          out[pos * 2 + 1] = (float)cls;
          ++pos; ++c2;
        }
      }
    }
  }
}

extern "C" void kernel_launch(void* const* d_in, const int* in_sizes, int n_in,
                              void* d_out, int out_size, void* d_ws, size_t ws_size,
                              hipStream_t stream) {
  const int*   y      = (const int*)d_in[0];     // setup_inputs order: y, scores
  const float* scores = (const float*)d_in[1];
  float*       out    = (float*)d_out;
  unsigned*    cm     = (unsigned*)d_ws;         // 400*400*4 = 640 KB scratch
  const int    batch  = in_sizes[0];

  const int cm_elems = NUM_CLASS * NUM_CLASS;
  zero_u32<<<(cm_elems + 255) / 256, 256, 0, stream>>>(cm, cm_elems);

  int ngroups = (batch + ROWS_PER_DMA - 1) / ROWS_PER_DMA;
  int blocks = 4096;
  int maxblocks = (ngroups + WAVES_PER_BLOCK - 1) / WAVES_PER_BLOCK;
  if (blocks > maxblocks) blocks = maxblocks;
  if (blocks < 1) blocks = 1;
  size_t shmem = (size_t)WAVES_PER_BLOCK * 2 * BUF_BYTES;   // 102400 B
  argmax_cm_kernel<<<blocks, THREADS_MAIN, shmem, stream>>>(scores, y, cm, batch);

  finalize_kernel<<<1, 512, 0, stream>>>(cm, out);
}